// DepthPrediction_12532714569769
// MI455X (gfx1250) — compile-verified
//
#include <hip/hip_runtime.h>
#include <hip/hip_bf16.h>
#include <math.h>

// Problem constants (fixed by the reference).
constexpr int NV = 3, NB = 2, NC = 16, NH = 128, NW = 128, ND = 32;
constexpr int HW = NH * NW;

typedef __attribute__((ext_vector_type(2))) float v2f;
typedef __attribute__((ext_vector_type(8))) float v8f;

// ---------------------------------------------------------------- helpers ---

__device__ inline void inv4(const float* m, float* out) {
    float inv[16];
    inv[0]  =  m[5]*m[10]*m[15] - m[5]*m[11]*m[14] - m[9]*m[6]*m[15] + m[9]*m[7]*m[14] + m[13]*m[6]*m[11] - m[13]*m[7]*m[10];
    inv[4]  = -m[4]*m[10]*m[15] + m[4]*m[11]*m[14] + m[8]*m[6]*m[15] - m[8]*m[7]*m[14] - m[12]*m[6]*m[11] + m[12]*m[7]*m[10];
    inv[8]  =  m[4]*m[9]*m[15]  - m[4]*m[11]*m[13] - m[8]*m[5]*m[15] + m[8]*m[7]*m[13] + m[12]*m[5]*m[11] - m[12]*m[7]*m[9];
    inv[12] = -m[4]*m[9]*m[14]  + m[4]*m[10]*m[13] + m[8]*m[5]*m[14] - m[8]*m[6]*m[13] - m[12]*m[5]*m[10] + m[12]*m[6]*m[9];
    inv[1]  = -m[1]*m[10]*m[15] + m[1]*m[11]*m[14] + m[9]*m[2]*m[15] - m[9]*m[3]*m[14] - m[13]*m[2]*m[11] + m[13]*m[3]*m[10];
    inv[5]  =  m[0]*m[10]*m[15] - m[0]*m[11]*m[14] - m[8]*m[2]*m[15] + m[8]*m[3]*m[14] + m[12]*m[2]*m[11] - m[12]*m[3]*m[10];
    inv[9]  = -m[0]*m[9]*m[15]  + m[0]*m[11]*m[13] + m[8]*m[1]*m[15] - m[8]*m[3]*m[13] - m[12]*m[1]*m[11] + m[12]*m[3]*m[9];
    inv[13] =  m[0]*m[9]*m[14]  - m[0]*m[10]*m[13] - m[8]*m[1]*m[14] + m[8]*m[2]*m[13] + m[12]*m[1]*m[10] - m[12]*m[2]*m[9];
    inv[2]  =  m[1]*m[6]*m[15]  - m[1]*m[7]*m[14]  - m[5]*m[2]*m[15] + m[5]*m[3]*m[14] + m[13]*m[2]*m[7]  - m[13]*m[3]*m[6];
    inv[6]  = -m[0]*m[6]*m[15]  + m[0]*m[7]*m[14]  + m[4]*m[2]*m[15] - m[4]*m[3]*m[14] - m[12]*m[2]*m[7]  + m[12]*m[3]*m[6];
    inv[10] =  m[0]*m[5]*m[15]  - m[0]*m[7]*m[13]  - m[4]*m[1]*m[15] + m[4]*m[3]*m[13] + m[12]*m[1]*m[7]  - m[12]*m[3]*m[5];
    inv[14] = -m[0]*m[5]*m[14]  + m[0]*m[6]*m[13]  + m[4]*m[1]*m[14] - m[4]*m[2]*m[13] - m[12]*m[1]*m[6]  + m[12]*m[2]*m[5];
    inv[3]  = -m[1]*m[6]*m[11]  + m[1]*m[7]*m[10]  + m[5]*m[2]*m[11] - m[5]*m[3]*m[10] - m[9]*m[2]*m[7]   + m[9]*m[3]*m[6];
    inv[7]  =  m[0]*m[6]*m[11]  - m[0]*m[7]*m[10]  - m[4]*m[2]*m[11] + m[4]*m[3]*m[10] + m[8]*m[2]*m[7]   - m[8]*m[3]*m[6];
    inv[11] = -m[0]*m[5]*m[11]  + m[0]*m[7]*m[9]   + m[4]*m[1]*m[11] - m[4]*m[3]*m[9]  - m[8]*m[1]*m[7]   + m[8]*m[3]*m[5];
    inv[15] =  m[0]*m[5]*m[10]  - m[0]*m[6]*m[9]   - m[4]*m[1]*m[10] + m[4]*m[2]*m[9]  + m[8]*m[1]*m[6]   - m[8]*m[2]*m[5];
    float det = m[0]*inv[0] + m[1]*inv[4] + m[2]*inv[8] + m[3]*inv[12];
    float id = 1.0f / det;
    #pragma unroll
    for (int k = 0; k < 16; ++k) out[k] = inv[k] * id;
}

__device__ inline void mul4(const float* a, const float* b, float* c) {
    #pragma unroll
    for (int r = 0; r < 4; ++r)
        #pragma unroll
        for (int cn = 0; cn < 4; ++cn) {
            float s = 0.f;
            #pragma unroll
            for (int k = 0; k < 4; ++k) s = fmaf(a[r*4+k], b[k*4+cn], s);
            c[r*4+cn] = s;
        }
}

__device__ inline float v8_at(v8f v, int i) {
    float r = v[0];
    r = (i == 1) ? v[1] : r;
    r = (i == 2) ? v[2] : r;
    r = (i == 3) ? v[3] : r;
    r = (i == 4) ? v[4] : r;
    r = (i == 5) ? v[5] : r;
    r = (i == 6) ? v[6] : r;
    r = (i == 7) ? v[7] : r;
    return r;
}

// --------------------------------------------------------------- kernel 1 ---
// Per (src view, batch): proj = src_KK @ src_w2c @ inv(ref_KK @ ref_w2c).
// Stores rot (row-major 3x3) + trans (3) = 12 floats at ws_proj[t*12].
__global__ void k_proj(const float* __restrict__ K, const float* __restrict__ c2w,
                       float* __restrict__ pr) {
    int t = threadIdx.x;
    if (t >= (NV - 1) * NB) return;
    int v = 1 + t / NB, b = t % NB;

    float sc2w[16], rc2w[16];
    #pragma unroll
    for (int k = 0; k < 16; ++k) { sc2w[k] = c2w[(v*NB + b)*16 + k]; rc2w[k] = c2w[b*16 + k]; }
    float sw2c[16], rw2c[16];
    inv4(sc2w, sw2c); inv4(rc2w, rw2c);

    float sKK[16], rKK[16];
    #pragma unroll
    for (int k = 0; k < 16; ++k) { sKK[k] = sw2c[k]; rKK[k] = rw2c[k]; }
    #pragma unroll
    for (int r = 0; r < 3; ++r)
        #pragma unroll
        for (int cn = 0; cn < 3; ++cn) {
            sKK[r*4+cn] = K[(v*NB + b)*9 + r*3 + cn];
            rKK[r*4+cn] = K[b*9 + r*3 + cn];
        }

    float t1[16], t2[16], t2i[16], P[16];
    mul4(sKK, sw2c, t1);
    mul4(rKK, rw2c, t2);
    inv4(t2, t2i);
    mul4(t1, t2i, P);

    float* o = pr + t * 12;
    o[0]=P[0]; o[1]=P[1]; o[2]=P[2];
    o[3]=P[4]; o[4]=P[5]; o[5]=P[6];
    o[6]=P[8]; o[7]=P[9]; o[8]=P[10];
    o[9]=P[3]; o[10]=P[7]; o[11]=P[11];
}

// --------------------------------------------------------------- kernel 2 ---
// w_feat[b][n][h][w] = safe_norm_C( f_nb - f_center ), zero-padded neighbors.
__global__ __launch_bounds__(256) void k_wfeat(const float* __restrict__ feat,
                                               float* __restrict__ wf) {
    int tid = blockIdx.x * 256 + threadIdx.x;          // < NB*25*HW
    int w = tid & 127, h = (tid >> 7) & 127;
    int n = (tid >> 14) % 25;
    int b = tid / (25 * HW);
    int i = n / 5, j = n % 5;
    int y = h + i - 2, x = w + j - 2;
    bool inb = ((unsigned)y < (unsigned)NH) && ((unsigned)x < (unsigned)NW);
    const float* fb = feat + (size_t)(b * NC) * HW;    // ref view (v=0)
    int coff = h * NW + w;
    int noff = inb ? (y * NW + x) : 0;
    float s = 0.f;
    #pragma unroll
    for (int c = 0; c < NC; ++c) {
        float fc = fb[c * HW + coff];
        float fn = inb ? fb[c * HW + noff] : 0.f;
        float d = fn - fc;
        s = fmaf(d, d, s);
    }
    wf[tid] = (s == 0.f) ? 0.f : sqrtf(s);
}

// --------------------------------------------------------------- kernel 3 ---
// Cost volume via homography warp + WMMA Gram-diagonal channel reduction.
// One wave per 16-pixel tile of one (b,d,h) row. Lanes 16..31 duplicate the
// pixels of lanes 0..15, which makes the A and B WMMA fragments identical.
__global__ __launch_bounds__(256) void k_cost(const float* __restrict__ feat,
                                              const float* __restrict__ depth,
                                              const float* __restrict__ proj,
                                              float* __restrict__ cost) {
    int gtid = blockIdx.x * 256 + threadIdx.x;
    int wave = gtid >> 5;
    int lane = threadIdx.x & 31;
    int tw = wave & 7;                 // W/16 = 8 tiles
    int h  = (wave >> 3) & 127;
    int d  = (wave >> 10) & 31;
    int b  = wave >> 15;
    int p  = lane & 15;
    int x  = tw * 16 + p;

    float dep = depth[((b * ND + d) * NH + h) * NW + x];

    const float* fref = feat + (size_t)(b * NC) * HW + h * NW + x;   // view 0
    __builtin_prefetch(fref, 0, 1);
    float frefv[NC];
    #pragma unroll
    for (int c = 0; c < NC; ++c) frefv[c] = fref[c * HW];

    float wsum[NC];
    #pragma unroll
    for (int c = 0; c < NC; ++c) wsum[c] = 0.f;

    bool hi = (lane >= 16);
    v8f s1 = {}, s2 = {};

    #pragma unroll
    for (int view = 1; view < NV; ++view) {
        const float* pr = proj + ((view - 1) * NB + b) * 12;
        float xf = (float)x, yf = (float)h;
        float rx = fmaf(pr[0], xf, fmaf(pr[1], yf, pr[2]));
        float ry = fmaf(pr[3], xf, fmaf(pr[4], yf, pr[5]));
        float rz = fmaf(pr[6], xf, fmaf(pr[7], yf, pr[8]));
        float px = fmaf(rx, dep, pr[9]);
        float py = fmaf(ry, dep, pr[10]);
        float pz = fmaf(rz, dep, pr[11]);
        float ix = px / pz, iy = py / pz;
        float gx = ix * (1.0f / ((NW - 1) * 0.5f)) - 1.0f;
        float gy = iy * (1.0f / ((NH - 1) * 0.5f)) - 1.0f;
        float gxp = (gx + 1.0f) * (NW * 0.5f) - 0.5f;
        float gyp = (gy + 1.0f) * (NH * 0.5f) - 0.5f;
        float x0f = floorf(gxp), y0f = floorf(gyp);
        float fx = gxp - x0f, fy = gyp - y0f;
        int x0 = (int)x0f, y0 = (int)y0f;

        const float* fv = feat + (size_t)((view * NB + b) * NC) * HW;
        __builtin_prefetch(fv, 0, 1);
        #pragma unroll
        for (int t = 0; t < 4; ++t) {
            int xt = x0 + (t & 1), yt = y0 + (t >> 1);
            float wt = ((t & 1) ? fx : 1.0f - fx) * ((t >> 1) ? fy : 1.0f - fy);
            bool valid = ((unsigned)xt < (unsigned)NW) && ((unsigned)yt < (unsigned)NH);
            wt = valid ? wt : 0.0f;
            int xc = min(max(xt, 0), NW - 1);
            int yc = min(max(yt, 0), NH - 1);
            const float* tp = fv + yc * NW + xc;
            #pragma unroll
            for (int c = 0; c < NC; ++c) wsum[c] = fmaf(wt, tp[c * HW], wsum[c]);
        }

        // diff = ref - cumsum(warped); Gram via WMMA: diag(D) = sum_c diff^2
        float diff[NC];
        #pragma unroll
        for (int c = 0; c < NC; ++c) diff[c] = frefv[c] - wsum[c];

        v8f acc = {};
        #pragma unroll
        for (int kc = 0; kc < 4; ++kc) {
            v2f a;
            a.x = hi ? diff[4*kc + 2] : diff[4*kc + 0];
            a.y = hi ? diff[4*kc + 3] : diff[4*kc + 1];
            // D = A x B + C, A == B fragment => D[m][n] = sum_k diff[m][k]*diff[n][k]
            acc = __builtin_amdgcn_wmma_f32_16x16x4_f32(false, a, false, a,
                                                        (short)0, acc, false, false);
        }
        if (view == 1) s1 = acc; else s2 = acc;
    }

    // Diagonal (pixel p, pixel p): lanes 0..7 own rows 0..7 (acc[lane]),
    // lanes 24..31 own rows 8..15 (acc[lane-24]).
    int r = (lane < 8) ? lane : ((lane >= 24) ? (lane - 24) : -1);
    if (r >= 0) {
        int pix = (lane < 8) ? lane : (lane - 16);
        float s = fminf(v8_at(s1, r), v8_at(s2, r));
        cost[((b * ND + d) * NH + h) * NW + tw * 16 + pix] = sqrtf(fmaxf(s, 0.0f));
    }
}

// --------------------------------------------------------------- kernel 4 ---
// agg[b,d,h,w] = sum_n cost_nb * softmax_n(|d_nb - d_c|) * w_feat[n]
__global__ __launch_bounds__(256) void k_agg(const float* __restrict__ depth,
                                             const float* __restrict__ cost,
                                             const float* __restrict__ wf,
                                             float* __restrict__ agg) {
    int tid = blockIdx.x * 256 + threadIdx.x;          // < NB*ND*HW
    int w = tid & 127, h = (tid >> 7) & 127;
    int d = (tid >> 14) & 31;
    int b = tid >> 19;
    const float* dep = depth + (size_t)(b * ND + d) * HW;
    const float* cv  = cost  + (size_t)(b * ND + d) * HW;
    const float* wfb = wf + (size_t)b * 25 * HW;
    int coff = h * NW + w;
    float dc = dep[coff];

    float a[25];
    float amax = 0.0f;                                  // |.| >= 0 and n=12 gives 0
    #pragma unroll
    for (int n = 0; n < 25; ++n) {
        int y = h + n / 5 - 2, x = w + n % 5 - 2;
        bool inb = ((unsigned)y < (unsigned)NH) && ((unsigned)x < (unsigned)NW);
        float dn = inb ? dep[y * NW + x] : 0.0f;
        float av = fabsf(dn - dc);
        a[n] = av;
        amax = fmaxf(amax, av);
    }
    float es = 0.0f, acc = 0.0f;
    #pragma unroll
    for (int n = 0; n < 25; ++n) {
        float e = expf(a[n] - amax);
        es += e;
        int y = h + n / 5 - 2, x = w + n % 5 - 2;
        bool inb = ((unsigned)y < (unsigned)NH) && ((unsigned)x < (unsigned)NW);
        float cn = inb ? cv[y * NW + x] : 0.0f;         // zero-padded cost
        float fn = wfb[n * HW + coff];                  // indexed at center pixel
        acc = fmaf(cn * fn, e, acc);
    }
    agg[tid] = acc / es;
}

// --------------------------------------------------------------- kernel 5 ---
// softmax over D + depth expectation -> out (B,H,W)
__global__ __launch_bounds__(256) void k_out(const float* __restrict__ agg,
                                             const float* __restrict__ depth,
                                             float* __restrict__ out) {
    int tid = blockIdx.x * 256 + threadIdx.x;           // < NB*HW
    int b = tid >> 14;
    int hw = tid & 16383;
    const float* ag  = agg   + (size_t)b * ND * HW;
    const float* dep = depth + (size_t)b * ND * HW;
    float a[ND];
    float m = -1e30f;
    #pragma unroll
    for (int d = 0; d < ND; ++d) { a[d] = ag[d * HW + hw]; m = fmaxf(m, a[d]); }
    float es = 0.0f, num = 0.0f;
    #pragma unroll
    for (int d = 0; d < ND; ++d) {
        float e = expf(a[d] - m);
        es += e;
        num = fmaf(e, dep[d * HW + hw], num);
    }
    out[tid] = num / es;
}

// ------------------------------------------------------------------ launch ---
extern "C" void kernel_launch(void* const* d_in, const int* in_sizes, int n_in,
                              void* d_out, int out_size, void* d_ws, size_t ws_size,
                              hipStream_t stream) {
    (void)in_sizes; (void)n_in; (void)out_size; (void)ws_size;
    const float* features   = (const float*)d_in[0];   // (V,B,C,H,W)
    const float* intrinsics = (const float*)d_in[1];   // (V,B,3,3)
    const float* c2w        = (const float*)d_in[2];   // (V,B,4,4)
    const float* depth      = (const float*)d_in[3];   // (B,D,H,W)
    float* out = (float*)d_out;                        // (B,H,W)

    float* ws       = (float*)d_ws;
    float* ws_proj  = ws;                              // 48 floats
    float* ws_wfeat = ws + 64;                         // NB*25*HW
    float* ws_cost  = ws_wfeat + (size_t)NB * 25 * HW; // NB*ND*HW
    float* ws_agg   = ws_cost  + (size_t)NB * ND * HW; // NB*ND*HW

    k_proj<<<1, 32, 0, stream>>>(intrinsics, c2w, ws_proj);
    k_wfeat<<<(NB * 25 * HW) / 256, 256, 0, stream>>>(features, ws_wfeat);
    // waves = NB*ND*NH*(NW/16) = 65536 ; 8 waves/block -> 8192 blocks
    k_cost<<<(NB * ND * NH * (NW / 16) * 32) / 256, 256, 0, stream>>>(features, depth, ws_proj, ws_cost);
    k_agg<<<(NB * ND * HW) / 256, 256, 0, stream>>>(depth, ws_cost, ws_wfeat, ws_agg);
    k_out<<<(NB * HW) / 256, 256, 0, stream>>>(ws_agg, depth, out);
}